// myGATConv_77068893159890
// MI455X (gfx1250) — compile-verified
//
#include <hip/hip_runtime.h>
#include <hip/hip_bf16.h>

typedef __attribute__((ext_vector_type(2))) float v2f;
typedef __attribute__((ext_vector_type(8))) float v8f;

#define HEADS 8
#define DIM 128
#define NEG_SLOPE 0.2f

// ---------------------------------------------------------------------------
// Order-preserving float <-> uint key for native u32 atomic max (segment_max).
// ---------------------------------------------------------------------------
__device__ __forceinline__ unsigned fkey(float f) {
    unsigned u = __float_as_uint(f);
    return (u & 0x80000000u) ? ~u : (u | 0x80000000u);
}
__device__ __forceinline__ float funkey(unsigned k) {
    return (k & 0x80000000u) ? __uint_as_float(k & 0x7fffffffu)
                             : __uint_as_float(~k);
}

// ---------------------------------------------------------------------------
// Fused QKV + residual projection: {q,k,v,res} = h @ {Wq,Wk,Wv,Wres}^T.
// One block per 16-row strip of h (staged ONCE in LDS, padded stride 132).
// 8 waves/block, one wave per 16-col output tile. Each wave keeps FOUR
// independent f32 accumulators and reuses each A fragment across the four
// weight matrices -> 4 independent WMMA chains per K step, 128 WMMAs/wave.
// ---------------------------------------------------------------------------
__global__ __launch_bounds__(256) void qkvr_gemm_wmma(
    const float* __restrict__ h,
    const float* __restrict__ Wq, const float* __restrict__ Wk,
    const float* __restrict__ Wv, const float* __restrict__ Wres,
    float* __restrict__ q, float* __restrict__ k, float* __restrict__ v,
    float* __restrict__ res, int N)
{
    __shared__ float As[16 * 132];

    const int mt   = blockIdx.x;
    const int t    = threadIdx.x;
    const int row0 = mt * 16;

    // Stage 16x128 strip of h into LDS (zero-fill past N).
    const float* hs = h + (size_t)row0 * DIM;
    for (int i = t; i < 16 * DIM; i += 256) {
        int r = i >> 7, c = i & 127;
        As[r * 132 + c] = (row0 + r < N) ? hs[i] : 0.0f;
    }
    __syncthreads();

    const int wave = t >> 5;        // 0..7 -> output column tile
    const int lane = t & 31;
    const int half = lane >> 4;     // 0: lanes 0-15, 1: lanes 16-31
    const int l    = lane & 15;

    // B fragment source: B[k][n] = W[n][k]; this lane owns column n = wave*16+l.
    const size_t wr = (size_t)(wave * 16 + l) * DIM;
    const float* Wr0 = Wq   + wr;
    const float* Wr1 = Wk   + wr;
    const float* Wr2 = Wv   + wr;
    const float* Wr3 = Wres + wr;
    const float* arow = As + l * 132;

    v8f cq = {}, ck = {}, cv = {}, cr = {};
#pragma unroll 4
    for (int kt = 0; kt < 32; ++kt) {
        const int kb = kt * 4 + 2 * half;   // K pair owned by this lane-half
        v2f a, b0, b1, b2, b3;
        a.x  = arow[kb];     a.y  = arow[kb + 1];
        b0.x = Wr0[kb];      b0.y = Wr0[kb + 1];
        b1.x = Wr1[kb];      b1.y = Wr1[kb + 1];
        b2.x = Wr2[kb];      b2.y = Wr2[kb + 1];
        b3.x = Wr3[kb];      b3.y = Wr3[kb + 1];
        cq = __builtin_amdgcn_wmma_f32_16x16x4_f32(false, a, false, b0, (short)0, cq, false, false);
        ck = __builtin_amdgcn_wmma_f32_16x16x4_f32(false, a, false, b1, (short)0, ck, false, false);
        cv = __builtin_amdgcn_wmma_f32_16x16x4_f32(false, a, false, b2, (short)0, cv, false, false);
        cr = __builtin_amdgcn_wmma_f32_16x16x4_f32(false, a, false, b3, (short)0, cr, false, false);
    }

    // C/D layout: VGPR r holds row M = r + 8*half, column N = l.
    const size_t obase = (size_t)(row0 + 8 * half) * DIM + wave * 16 + l;
    float* oq = q   + obase;
    float* ok = k   + obase;
    float* ov = v   + obase;
    float* orr = res + obase;
    if (row0 + 16 <= N) {           // full tile: straight-line stores
#pragma unroll
        for (int r = 0; r < 8; ++r) {
            oq[(size_t)r * DIM]  = cq[r];
            ok[(size_t)r * DIM]  = ck[r];
            ov[(size_t)r * DIM]  = cv[r];
            orr[(size_t)r * DIM] = cr[r];
        }
    } else {                        // ragged tail tile
#pragma unroll
        for (int r = 0; r < 8; ++r) {
            if (row0 + 8 * half + r < N) {
                oq[(size_t)r * DIM]  = cq[r];
                ok[(size_t)r * DIM]  = ck[r];
                ov[(size_t)r * DIM]  = cv[r];
                orr[(size_t)r * DIM] = cr[r];
            }
        }
    }
}

// ---------------------------------------------------------------------------
// Init segment-max keys and softmax denominators.
// ---------------------------------------------------------------------------
__global__ void init_mz(unsigned* __restrict__ mkey, float* __restrict__ z, int NH)
{
    int i = blockIdx.x * blockDim.x + threadIdx.x;
    if (i < NH) { mkey[i] = 0u; z[i] = 0.0f; }
}

// ---------------------------------------------------------------------------
// Per-(edge,head) attention score + leaky relu + segment max (atomic u32 max).
// ---------------------------------------------------------------------------
__global__ void edge_scores(const float* __restrict__ q, const float* __restrict__ k,
                            const int* __restrict__ src, const int* __restrict__ dst,
                            float* __restrict__ e, unsigned* __restrict__ mkey,
                            long long EH)
{
    long long idx = (long long)blockIdx.x * blockDim.x + threadIdx.x;
    if (idx >= EH) return;
    const int eid = (int)(idx >> 3);
    const int hd  = (int)(idx & 7);
    const int s = src[eid], d = dst[eid];

    const float4* qp = (const float4*)(q + (size_t)d * DIM + hd * 16);
    const float4* kp = (const float4*)(k + (size_t)s * DIM + hd * 16);
    float acc = 0.0f;
#pragma unroll
    for (int i = 0; i < 4; ++i) {
        float4 a = qp[i], b = kp[i];
        acc += a.x * b.x + a.y * b.y + a.z * b.z + a.w * b.w;
    }
    acc = (acc >= 0.0f) ? acc : NEG_SLOPE * acc;
    e[idx] = acc;
    atomicMax(&mkey[(size_t)d * HEADS + hd], fkey(acc));
}

// ---------------------------------------------------------------------------
// ex = exp(e - m[dst]); z[dst] += ex (atomic f32 add). ex overwrites e in place.
// ---------------------------------------------------------------------------
__global__ void edge_exp(const int* __restrict__ dst, float* __restrict__ e,
                         const unsigned* __restrict__ mkey, float* __restrict__ z,
                         long long EH)
{
    long long idx = (long long)blockIdx.x * blockDim.x + threadIdx.x;
    if (idx >= EH) return;
    const int eid = (int)(idx >> 3);
    const int hd  = (int)(idx & 7);
    const int d = dst[eid];
    const float m  = funkey(mkey[(size_t)d * HEADS + hd]);
    const float ex = __expf(e[idx] - m);
    e[idx] = ex;
    atomicAdd(&z[(size_t)d * HEADS + hd], ex);
}

// ---------------------------------------------------------------------------
// out[dst] += (ex/z[dst]) * v[src]   (16 f32 atomic adds per (edge,head)).
// out already holds the residual h@Wres^T from the GEMM stage.
// ---------------------------------------------------------------------------
__global__ void edge_aggregate(const int* __restrict__ src, const int* __restrict__ dst,
                               const float* __restrict__ e, const float* __restrict__ z,
                               const float* __restrict__ v, float* __restrict__ out,
                               long long EH)
{
    long long idx = (long long)blockIdx.x * blockDim.x + threadIdx.x;
    if (idx >= EH) return;
    const int eid = (int)(idx >> 3);
    const int hd  = (int)(idx & 7);
    const int s = src[eid], d = dst[eid];
    const float alpha = e[idx] / z[(size_t)d * HEADS + hd];

    const float* vp = v   + (size_t)s * DIM + hd * 16;
    float*       op = out + (size_t)d * DIM + hd * 16;
#pragma unroll
    for (int i = 0; i < 16; ++i) atomicAdd(&op[i], alpha * vp[i]);
}

// ---------------------------------------------------------------------------
// Launch. Inputs (setup_inputs order):
//  0: h [N,128]  1: edge_features (dead)  2: src [E]  3: dst [E]
//  4: Wq  5: Wk  6: Wv  7: We (dead)  8: Wres
// ---------------------------------------------------------------------------
extern "C" void kernel_launch(void* const* d_in, const int* in_sizes, int n_in,
                              void* d_out, int out_size, void* d_ws, size_t ws_size,
                              hipStream_t stream)
{
    const float* h    = (const float*)d_in[0];
    const int*   src  = (const int*)d_in[2];
    const int*   dst  = (const int*)d_in[3];
    const float* Wq   = (const float*)d_in[4];
    const float* Wk   = (const float*)d_in[5];
    const float* Wv   = (const float*)d_in[6];
    const float* Wres = (const float*)d_in[8];
    float* out = (float*)d_out;

    const int N = in_sizes[0] / DIM;
    const int E = in_sizes[2];

    // Workspace carve-up (~106 MB): q, k, v, e[E,H], mkey[N,H], z[N,H]
    float* q = (float*)d_ws;
    float* k = q + (size_t)N * DIM;
    float* v = k + (size_t)N * DIM;
    float* e = v + (size_t)N * DIM;
    unsigned* mkey = (unsigned*)(e + (size_t)E * HEADS);
    float* z = (float*)(mkey + (size_t)N * HEADS);

    // 1) Fused QKV + residual projections via f32 WMMA (residual -> d_out).
    qkvr_gemm_wmma<<<(N + 15) / 16, 256, 0, stream>>>(h, Wq, Wk, Wv, Wres,
                                                      q, k, v, out, N);

    // 2) Init softmax state.
    const int NH = N * HEADS;
    init_mz<<<(NH + 255) / 256, 256, 0, stream>>>(mkey, z, NH);

    // 3..5) Edge pipeline.
    const long long EH = (long long)E * HEADS;
    const int eblocks = (int)((EH + 255) / 256);
    edge_scores<<<eblocks, 256, 0, stream>>>(q, k, src, dst, e, mkey, EH);
    edge_exp<<<eblocks, 256, 0, stream>>>(dst, e, mkey, z, EH);
    edge_aggregate<<<eblocks, 256, 0, stream>>>(src, dst, e, z, v, out, EH);
}